// HausdorffDTLoss_34213709480652
// MI455X (gfx1250) — compile-verified
//
#include <hip/hip_runtime.h>
#include <math.h>

// Problem constants (from reference): B=4, H=W=192, ALPHA=2
#define Bn   4
#define Hh   192
#define Ww   192
#define NIMG 8   // 4 pred images + 4 target images

typedef __attribute__((ext_vector_type(2))) float v2f;
typedef __attribute__((ext_vector_type(8))) float v8f;

// ---------------------------------------------------------------------------
// Zero the per-image "has foreground" flags (graph-capture-safe init).
// ---------------------------------------------------------------------------
__global__ void hdt_init_flags(int* __restrict__ flags) {
    const int t = threadIdx.x;
    if (t < NIMG) flags[t] = 0;
}

// ---------------------------------------------------------------------------
// Column pass: for image im, row r, compute
//   mFG[im][r][y] = min over j with mask[r][j]==0 of (y-j)^2   (for edt(fg))
//   mBG[im][r][y] = min over j with mask[r][j]==1 of (y-j)^2   (for edt(~fg))
// One block per (image,row); thread y scans the LDS-staged mask row.
// ---------------------------------------------------------------------------
__global__ void hdt_colpass(const float* __restrict__ pred,
                            const float* __restrict__ target,
                            float* __restrict__ mFG,
                            float* __restrict__ mBG,
                            int*   __restrict__ flags) {
    __shared__ int s_mask[Ww];
    __shared__ int s_any;
    const int blk = blockIdx.x;
    const int im  = blk / Hh;
    const int r   = blk % Hh;
    const int y   = threadIdx.x;

    const float* img = (im < Bn) ? (pred + (size_t)im * Hh * Ww)
                                 : (target + (size_t)(im - Bn) * Hh * Ww);
    if (y == 0) s_any = 0;
    __syncthreads();

    const int fg = (img[r * Ww + y] > 0.5f) ? 1 : 0;
    s_mask[y] = fg;
    if (fg) s_any = 1;               // benign same-value race
    __syncthreads();

    float mfg = __builtin_inff();    // min over background columns
    float mbg = __builtin_inff();    // min over foreground columns
#pragma unroll 4
    for (int j = 0; j < Ww; ++j) {
        const int   d  = y - j;
        const float d2 = (float)(d * d);       // exact in f32 (<= 191^2)
        if (s_mask[j]) mbg = fminf(mbg, d2);
        else           mfg = fminf(mfg, d2);
    }
    const size_t o = (size_t)(im * Hh + r) * Ww + y;
    mFG[o] = mfg;
    mBG[o] = mbg;
    if (y == 0 && s_any) atomicOr(&flags[im], 1);   // idempotent -> deterministic
}

// ---------------------------------------------------------------------------
// Row pass + pointwise combine. One block per (batch b, row x); thread y.
//   d2*(x,y) = min_i (x-i)^2 + m*[i][y]      (min-plus envelope)
// Select fg/bg result by each image's own mask bit (sqrt cancels: ALPHA==2),
// apply the any(fg) guard, multiply by (pred-target)^2, and produce a
// deterministic per-block partial sum.
// ---------------------------------------------------------------------------
__global__ void hdt_rowpass(const float* __restrict__ pred,
                            const float* __restrict__ target,
                            const float* __restrict__ mFG,
                            const float* __restrict__ mBG,
                            const int*   __restrict__ flags,
                            float* __restrict__ partials) {
    const int blk = blockIdx.x;          // b*Hh + x
    const int b   = blk / Hh;
    const int x   = blk % Hh;
    const int y   = threadIdx.x;

    const size_t slab = (size_t)Hh * Ww;
    const float* mFGp = mFG + (size_t)b * slab;          // pred image b
    const float* mBGp = mBG + (size_t)b * slab;
    const float* mFGt = mFG + (size_t)(Bn + b) * slab;   // target image b
    const float* mBGt = mBG + (size_t)(Bn + b) * slab;

    float dpf = __builtin_inff(), dpb = __builtin_inff();
    float dtf = __builtin_inff(), dtb = __builtin_inff();
#pragma unroll 2
    for (int i = 0; i < Hh; ++i) {
        const int   dx  = x - i;
        const float dx2 = (float)(dx * dx);
        const int   o   = i * Ww + y;                    // coalesced across y
        dpf = fminf(dpf, dx2 + mFGp[o]);
        dpb = fminf(dpb, dx2 + mBGp[o]);
        dtf = fminf(dtf, dx2 + mFGt[o]);
        dtb = fminf(dtb, dx2 + mBGt[o]);
    }

    const size_t po = (size_t)(b * Hh + x) * Ww + y;
    const float pv = pred[po];
    const float tv = target[po];
    const float d2p = (pv > 0.5f) ? dpf : dpb;   // pred_dt^2 at this pixel
    const float d2t = (tv > 0.5f) ? dtf : dtb;   // target_dt^2 at this pixel
    const float dist = (flags[b]      ? d2p : 0.0f)
                     + (flags[Bn + b] ? d2t : 0.0f);
    const float err  = (pv - tv) * (pv - tv);
    float val = err * dist;

    // Deterministic block reduction: per-wave shuffle tree, then serial 6-way.
    __shared__ float s_w[Ww / 32];
    float v = val;
    for (int off = 16; off > 0; off >>= 1) v += __shfl_down(v, off);  // wave32
    const int lane = y & 31, wid = y >> 5;
    if (lane == 0) s_w[wid] = v;
    __syncthreads();
    if (y == 0) {
        float s = 0.0f;
        for (int w = 0; w < Ww / 32; ++w) s += s_w[w];
        partials[blk] = s;
    }
}

// ---------------------------------------------------------------------------
// Final mean via V_WMMA_F32_16X16X4_F32: sum 768 partials as 12 chunks of 64.
// A = 16x4 f32 data chunk, B = 4x16 ones -> D[r][*] += row-sums, accumulated
// in the f32 C matrix (full precision, fixed hardware order -> deterministic).
// Single wave32, no divergence before the WMMAs (EXEC must be all 1s).
// A layout (ISA 7.12.2, 32-bit A 16x4): lane<16 holds A[lane][0..1] in v0..v1,
// lane>=16 holds A[lane-16][2..3]. We map chunk value c[m*4+k] = A[m][k].
// C layout: lanes 0..15 hold M=0..7 (v0..v7), lanes 16..31 hold M=8..15.
// ---------------------------------------------------------------------------
__global__ void hdt_reduce_wmma(const float* __restrict__ partials,
                                float* __restrict__ out) {
    const int lane = threadIdx.x;             // blockDim.x == 32
    const int row  = lane & 15;
    const int koff = (lane >> 4) << 1;        // 0 for lanes 0..15, 2 for 16..31

    v8f c = {};
    v2f ones; ones.x = 1.0f; ones.y = 1.0f;   // all-ones B: layout-agnostic

    for (int chunk = 0; chunk < (Bn * Hh) / 64; ++chunk) {   // 12 chunks
        const int base = chunk * 64 + row * 4 + koff;
        v2f a; a.x = partials[base]; a.y = partials[base + 1];
        c = __builtin_amdgcn_wmma_f32_16x16x4_f32(
                /*neg_a=*/false, a, /*neg_b=*/false, ones,
                /*c_mod=*/(short)0, c, /*reuse_a=*/false, /*reuse_b=*/false);
    }
    // Row-sums are replicated across N; lanes<16 carry M=0..7, lanes>=16 M=8..15.
    float s = c[0] + c[1] + c[2] + c[3] + c[4] + c[5] + c[6] + c[7];
    s += __shfl_xor(s, 16);
    if (lane == 0) out[0] = s / (float)(Bn * Hh * Ww);
}

// ---------------------------------------------------------------------------
extern "C" void kernel_launch(void* const* d_in, const int* in_sizes, int n_in,
                              void* d_out, int out_size, void* d_ws, size_t ws_size,
                              hipStream_t stream) {
    (void)in_sizes; (void)n_in; (void)out_size; (void)ws_size;

    const float* pred   = (const float*)d_in[0];
    const float* target = (const float*)d_in[1];
    float* out = (float*)d_out;

    // Workspace carve-up (~2.4 MB total; all L2-resident):
    float* mFG      = (float*)d_ws;                       // NIMG*H*W f32
    float* mBG      = mFG + (size_t)NIMG * Hh * Ww;       // NIMG*H*W f32
    float* partials = mBG + (size_t)NIMG * Hh * Ww;       // Bn*H f32 (=768)
    int*   flags    = (int*)(partials + (size_t)Bn * Hh); // NIMG ints

    hdt_init_flags<<<1, 32, 0, stream>>>(flags);
    hdt_colpass  <<<NIMG * Hh, Ww, 0, stream>>>(pred, target, mFG, mBG, flags);
    hdt_rowpass  <<<Bn * Hh,   Ww, 0, stream>>>(pred, target, mFG, mBG, flags, partials);
    hdt_reduce_wmma<<<1, 32, 0, stream>>>(partials, out);
}